// GCNEncoder_31774168056042
// MI455X (gfx1250) — compile-verified
//
#include <hip/hip_runtime.h>

typedef __attribute__((ext_vector_type(2))) float v2f;
typedef __attribute__((ext_vector_type(8))) float v8f;

#define DIM 128

__device__ __forceinline__ void atomAddF(float* p, float v) {
    unsafeAtomicAdd(p, v);   // hardware global_atomic_add_f32
}

// ---------------- degree / norm ----------------
__global__ void k_deg_init(float* __restrict__ deg, int n) {
    int i = blockIdx.x * blockDim.x + threadIdx.x;
    if (i < n) deg[i] = 1.0f;              // self loop
}

__global__ void k_deg_edges(const long long* __restrict__ ei, long long E,
                            float* __restrict__ deg) {
    long long e = (long long)blockIdx.x * blockDim.x + threadIdx.x;
    if (e < E) atomAddF(&deg[ei[E + e]], 1.0f);   // dst row of edge_index
}

__global__ void k_rsqrt_inplace(float* __restrict__ deg, int n) {
    int i = blockIdx.x * blockDim.x + threadIdx.x;
    if (i < n) deg[i] = rsqrtf(deg[i]);    // deg >= 1 always
}

// ---------------- WMMA f32 GEMM: H = Z (n x 128) * W (128 x 128) ----------------
// Block = 256 threads = 8 waves; block handles 16 rows, each wave one 16-col tile.
__global__ __launch_bounds__(256) void k_gemm_wmma(const float* __restrict__ Z,
                                                   const float* __restrict__ W,
                                                   float* __restrict__ H, int n) {
    __shared__ float zt[16 * DIM];         // 8 KB tile of Z
    int m0 = blockIdx.x * 16;
    int rows = n - m0; if (rows > 16) rows = 16;

    if (rows == 16) {
        // stage 16x128 Z tile into LDS (2048 floats = 512 float4)
        const float4* zsrc = (const float4*)(Z + (size_t)m0 * DIM);
        float4* zt4 = (float4*)zt;
        zt4[threadIdx.x]       = zsrc[threadIdx.x];
        zt4[threadIdx.x + 256] = zsrc[threadIdx.x + 256];
        __syncthreads();

        int wave = threadIdx.x >> 5;
        int lane = threadIdx.x & 31;
        int half = lane >> 4;              // 0: K=0,1   1: K=2,3
        int l16  = lane & 15;
        int n0   = wave * 16;

        v8f acc = {};
        #pragma unroll
        for (int k0 = 0; k0 < DIM; k0 += 4) {
            // A fragment (16x4 f32): lane row = l16, VGPR0=K(2*half), VGPR1=K(2*half+1)
            v2f a = *(const v2f*)&zt[l16 * DIM + k0 + 2 * half];
            // B fragment (4x16 f32): lane col = n0+l16, same K split
            v2f b;
            b.x = W[(k0 + 2 * half)     * DIM + n0 + l16];
            b.y = W[(k0 + 2 * half + 1) * DIM + n0 + l16];
            acc = __builtin_amdgcn_wmma_f32_16x16x4_f32(
                false, a, false, b, (short)0, acc, false, false);
        }
        // D layout: VGPR j -> row m0 + j + 8*half, col n0 + l16
        #pragma unroll
        for (int j = 0; j < 8; ++j) {
            int row = m0 + j + 8 * half;
            H[(size_t)row * DIM + n0 + l16] = acc[j];
        }
    } else {
        // scalar tail (block-uniform branch keeps WMMA path EXEC=all-1s)
        for (int r = 0; r < rows; ++r) {
            const float* zr = Z + (size_t)(m0 + r) * DIM;
            for (int c = threadIdx.x; c < DIM; c += blockDim.x) {
                float s = 0.f;
                for (int k = 0; k < DIM; ++k) s = fmaf(zr[k], W[k * DIM + c], s);
                H[(size_t)(m0 + r) * DIM + c] = s;
            }
        }
    }
}

// ---------------- self-loop init: agg = dis[row]^2 * H ; zero stats ----------------
__global__ void k_agg_init(const float4* __restrict__ H4, const float* __restrict__ dis,
                           float4* __restrict__ A4, float* __restrict__ stats,
                           long long nv4) {
    if (blockIdx.x == 0 && threadIdx.x < 256) stats[threadIdx.x] = 0.f;
    long long stride = (long long)gridDim.x * blockDim.x;
    for (long long i = (long long)blockIdx.x * blockDim.x + threadIdx.x; i < nv4; i += stride) {
        int row = (int)(i >> 5);           // 32 float4 per row
        float c = dis[row]; c = c * c;
        float4 v = H4[i];
        v.x *= c; v.y *= c; v.z *= c; v.w *= c;
        A4[i] = v;
    }
}

// ---------------- edge scatter: one wave32 per edge ----------------
__global__ __launch_bounds__(256) void k_edge_agg(const long long* __restrict__ ei, long long E,
                                                  const float* __restrict__ dis,
                                                  const float* __restrict__ H,
                                                  float* __restrict__ A) {
    long long wid = (long long)blockIdx.x * 8 + (threadIdx.x >> 5);
    int lane = threadIdx.x & 31;
    if (wid >= E) return;
    long long s = ei[wid];
    long long d = ei[E + wid];
    float c = dis[s] * dis[d];
    float4 v = ((const float4*)(H + s * DIM))[lane];   // 512B coalesced gather
    float* ap = A + d * DIM + lane * 4;
    atomAddF(ap + 0, c * v.x);
    atomAddF(ap + 1, c * v.y);
    atomAddF(ap + 2, c * v.z);
    atomAddF(ap + 3, c * v.w);
}

// ---------------- bias + ReLU + column stats ----------------
__global__ __launch_bounds__(256) void k_relu_stats(const float* __restrict__ A,
                                                    const float* __restrict__ bias,
                                                    float* __restrict__ Y,
                                                    float* __restrict__ stats, int n) {
    int col = threadIdx.x & 127;
    int rb  = threadIdx.x >> 7;            // 2 rows per block iteration
    float b = bias[col];
    float s = 0.f, sq = 0.f;
    for (int r = blockIdx.x * 2 + rb; r < n; r += gridDim.x * 2) {
        float v = A[(size_t)r * DIM + col] + b;
        v = fmaxf(v, 0.f);
        Y[(size_t)r * DIM + col] = v;
        s += v; sq += v * v;
    }
    atomAddF(&stats[col], s);
    atomAddF(&stats[128 + col], sq);
}

// ---------------- fold stats -> scale/shift ----------------
__global__ void k_bn_param(float* __restrict__ stats, const float* __restrict__ g,
                           const float* __restrict__ be, float invN) {
    int c = threadIdx.x;                   // 128 threads
    float mean = stats[c] * invN;
    float var  = stats[128 + c] * invN - mean * mean;
    float inv  = rsqrtf(var + 1e-5f);
    float sc   = g[c] * inv;
    stats[c]       = sc;
    stats[128 + c] = be[c] - mean * sc;
}

// ---------------- apply affine BN ----------------
__global__ __launch_bounds__(256) void k_bn_apply(const float4* __restrict__ Y4,
                                                  const float* __restrict__ stats,
                                                  float4* __restrict__ O4, long long nv4) {
    long long stride = (long long)gridDim.x * blockDim.x;
    for (long long i = (long long)blockIdx.x * blockDim.x + threadIdx.x; i < nv4; i += stride) {
        int c4 = ((int)i & 31) * 4;
        float4 v = Y4[i];
        v.x = v.x * stats[c4 + 0] + stats[128 + c4 + 0];
        v.y = v.y * stats[c4 + 1] + stats[128 + c4 + 1];
        v.z = v.z * stats[c4 + 2] + stats[128 + c4 + 2];
        v.w = v.w * stats[c4 + 3] + stats[128 + c4 + 3];
        O4[i] = v;
    }
}

extern "C" void kernel_launch(void* const* d_in, const int* in_sizes, int n_in,
                              void* d_out, int out_size, void* d_ws, size_t ws_size,
                              hipStream_t stream) {
    const float*     x  = (const float*)d_in[0];
    const long long* ei = (const long long*)d_in[1];
    const float* Wl[3] = {(const float*)d_in[2], (const float*)d_in[6],  (const float*)d_in[10]};
    const float* bl[3] = {(const float*)d_in[3], (const float*)d_in[7],  (const float*)d_in[11]};
    const float* gl[3] = {(const float*)d_in[4], (const float*)d_in[8],  (const float*)d_in[12]};
    const float* el[3] = {(const float*)d_in[5], (const float*)d_in[9],  (const float*)d_in[13]};

    int n = in_sizes[0] / DIM;             // 100000
    long long E = (long long)in_sizes[1] / 2;

    // workspace carve
    float* dis   = (float*)d_ws;                                   // n
    float* zbuf  = dis  + (((size_t)n + 255) & ~(size_t)255);      // n*DIM
    float* H     = zbuf + (size_t)n * DIM;                         // n*DIM
    float* A     = H    + (size_t)n * DIM;                         // n*DIM
    float* stats = A    + (size_t)n * DIM;                         // 256

    // degrees -> dis
    k_deg_init<<<(n + 255) / 256, 256, 0, stream>>>(dis, n);
    k_deg_edges<<<(unsigned)((E + 255) / 256), 256, 0, stream>>>(ei, E, dis);
    k_rsqrt_inplace<<<(n + 255) / 256, 256, 0, stream>>>(dis, n);

    int rowTiles = (n + 15) / 16;
    long long nv4 = (long long)n * (DIM / 4);
    const float* zin = x;
    for (int l = 0; l < 3; ++l) {
        k_gemm_wmma<<<rowTiles, 256, 0, stream>>>(zin, Wl[l], H, n);
        k_agg_init<<<2048, 256, 0, stream>>>((const float4*)H, dis, (float4*)A, stats, nv4);
        k_edge_agg<<<(unsigned)((E + 7) / 8), 256, 0, stream>>>(ei, E, dis, H, A);
        k_relu_stats<<<1024, 256, 0, stream>>>(A, bl[l], H, stats, n);
        k_bn_param<<<1, 128, 0, stream>>>(stats, gl[l], el[l], 1.0f / (float)n);
        float* outp = (l == 2) ? (float*)d_out : zbuf;
        k_bn_apply<<<4096, 256, 0, stream>>>((const float4*)H, stats, (float4*)outp, nv4);
        zin = zbuf;
    }
}